// ElevateH1toH2_36532991820311
// MI455X (gfx1250) — compile-verified
//
#include <hip/hip_runtime.h>
#include <math.h>

// Problem constants (reference: B=8, S=2048, D=256, F=2D=512)
constexpr int NB = 8;
constexpr int NS = 2048;
constexpr int ND = 256;
constexpr int NF = 512;

typedef float v2f __attribute__((ext_vector_type(2)));
typedef float v8f __attribute__((ext_vector_type(8)));

// ---------------------------------------------------------------------------
// Kernel 1: row L2-normalize  nrm = x / max(||x||, 1e-12)
// One block (256 threads) per row, D == 256.
// ---------------------------------------------------------------------------
__global__ __launch_bounds__(256) void normalize_kernel(
    const float* __restrict__ x, float* __restrict__ nrm) {
  const int r = blockIdx.x;
  const int t = threadIdx.x;
  const float v = x[(size_t)r * ND + t];
  __shared__ float red[256];
  red[t] = v * v;
  __syncthreads();
  for (int off = 128; off > 0; off >>= 1) {
    if (t < off) red[t] += red[t + off];
    __syncthreads();
  }
  const float denom = fmaxf(sqrtf(red[0]), 1e-12f);
  nrm[(size_t)r * ND + t] = v / denom;
}

// ---------------------------------------------------------------------------
// Kernel 2: sim = nrm_b * nrm_b^T  (one batch), fp32 WMMA 16x16x4.
// Block = 8 waves. The block's 32-row A-panel (shared by all waves) is staged
// in LDS (stride 260 floats -> conflict-free 16-lane fragment reads).
// Each wave computes a 32x32 output region = 4 accumulator tiles:
//   per k-step: 2 LDS b64 (A0,A1) + 2 global b64 (B0,B1) + 4 WMMAs
//   -> 8 FLOP/byte and 4 independent WMMA accumulation chains.
// Fragment pattern (16x16x4 f32): lane = m + 16*h holds float2 of row
// (tile+m) at column k+2h for both A and B (problem is symmetric).
// ---------------------------------------------------------------------------
__global__ __launch_bounds__(256) void sim_kernel(
    const float* __restrict__ nrm_b, float* __restrict__ sim) {
  __shared__ __align__(16) float apan[32 * 260];  // 33280 B
  const int tid = threadIdx.x;
  const int wave = tid >> 5;
  const int lane = tid & 31;
  const int lrow = lane & 15;
  const int half = lane >> 4;
  const int i0 = blockIdx.x * 32;
  const int j0 = (blockIdx.y * 8 + wave) * 32;

  // stage A panel: rows i0 .. i0+31, all 256 columns (coalesced)
  for (int idx = tid; idx < 32 * ND; idx += 256) {
    const int rr = idx >> 8;       // / 256
    const int c = idx & (ND - 1);  // % 256
    apan[rr * 260 + c] = nrm_b[(size_t)(i0 + rr) * ND + c];
  }
  __syncthreads();

  const float* a0 = apan + lrow * 260 + 2 * half;
  const float* a1 = apan + (16 + lrow) * 260 + 2 * half;
  const float* b0 = nrm_b + (size_t)(j0 + lrow) * ND + 2 * half;
  const float* b1 = nrm_b + (size_t)(j0 + 16 + lrow) * ND + 2 * half;

  v8f acc00 = {}, acc01 = {}, acc10 = {}, acc11 = {};
  for (int k = 0; k < ND; k += 4) {
    v2f fa0 = *(const v2f*)(a0 + k);  // ds_load_b64
    v2f fa1 = *(const v2f*)(a1 + k);
    v2f fb0 = *(const v2f*)(b0 + k);  // global_load_b64
    v2f fb1 = *(const v2f*)(b1 + k);
    acc00 = __builtin_amdgcn_wmma_f32_16x16x4_f32(
        false, fa0, false, fb0, (short)0, acc00, false, false);
    acc01 = __builtin_amdgcn_wmma_f32_16x16x4_f32(
        false, fa0, false, fb1, (short)0, acc01, false, false);
    acc10 = __builtin_amdgcn_wmma_f32_16x16x4_f32(
        false, fa1, false, fb0, (short)0, acc10, false, false);
    acc11 = __builtin_amdgcn_wmma_f32_16x16x4_f32(
        false, fa1, false, fb1, (short)0, acc11, false, false);
  }

#pragma unroll
  for (int v = 0; v < 8; ++v) {
    const int mi = v + 8 * half;
    float* r0 = sim + (size_t)(i0 + mi) * NS;
    float* r1 = sim + (size_t)(i0 + 16 + mi) * NS;
    r0[j0 + lrow] = acc00[v];
    r0[j0 + 16 + lrow] = acc01[v];
    r1[j0 + lrow] = acc10[v];
    r1[j0 + 16 + lrow] = acc11[v];
  }
}

// ---------------------------------------------------------------------------
// Kernel 3: per-row top-4 (k+1 = 4, self included) over sim row.
// One thread per row; float4 streaming scan; sim row is L2-resident.
// ---------------------------------------------------------------------------
__global__ __launch_bounds__(256) void topk_kernel(
    const float* __restrict__ sim, float* __restrict__ topv,
    int* __restrict__ topi, int rowbase) {
  const int row = blockIdx.x * 256 + threadIdx.x;
  const float4* p = (const float4*)(sim + (size_t)row * NS);
  float tv0 = -1e30f, tv1 = -1e30f, tv2 = -1e30f, tv3 = -1e30f;
  int ti0 = 0, ti1 = 0, ti2 = 0, ti3 = 0;

#define INS4(val, jj)                                                   \
  do {                                                                  \
    const float _v = (val);                                             \
    const int _j = (jj);                                                \
    if (_v > tv3) {                                                     \
      if (_v > tv2) {                                                   \
        tv3 = tv2; ti3 = ti2;                                           \
        if (_v > tv1) {                                                 \
          tv2 = tv1; ti2 = ti1;                                         \
          if (_v > tv0) {                                                \
            tv1 = tv0; ti1 = ti0; tv0 = _v; ti0 = _j;                   \
          } else { tv1 = _v; ti1 = _j; }                                \
        } else { tv2 = _v; ti2 = _j; }                                  \
      } else { tv3 = _v; ti3 = _j; }                                    \
    }                                                                   \
  } while (0)

  for (int q = 0; q < NS / 4; ++q) {
    const float4 f = p[q];
    INS4(f.x, 4 * q + 0);
    INS4(f.y, 4 * q + 1);
    INS4(f.z, 4 * q + 2);
    INS4(f.w, 4 * q + 3);
  }
#undef INS4
  const size_t o = (size_t)(rowbase + row) * 4;
  topv[o + 0] = tv0; topv[o + 1] = tv1; topv[o + 2] = tv2; topv[o + 3] = tv3;
  topi[o + 0] = ti0; topi[o + 1] = ti1; topi[o + 2] = ti2; topi[o + 3] = ti3;
}

// ---------------------------------------------------------------------------
// Kernel 4 (gather): nbr[i] = sum_{j in top(i), j!=i} sim_ij * x[j];
// cnt[i] = sum of those sim values.  Initializes nbr/cnt (plain stores).
// ---------------------------------------------------------------------------
__global__ __launch_bounds__(256) void gather_kernel(
    const float* __restrict__ x, const float* __restrict__ topv,
    const int* __restrict__ topi, float* __restrict__ nbr,
    float* __restrict__ cnt) {
  const int r = blockIdx.x;          // global row in [0, B*S)
  const int b = r / NS;
  const int i = r - b * NS;
  const int d = threadIdx.x;
  float sum = 0.f, c = 0.f;
#pragma unroll
  for (int s = 0; s < 4; ++s) {
    const int j = topi[(size_t)r * 4 + s];
    if (j == i) continue;
    const float v = topv[(size_t)r * 4 + s];
    sum += v * x[((size_t)b * NS + j) * ND + d];
    c += v;
  }
  nbr[(size_t)r * ND + d] = sum;
  if (d == 0) cnt[r] = c;
}

// ---------------------------------------------------------------------------
// Kernel 5 (scatter): symmetric completion.  For each directed edge j->i
// (i in top(j), i != j) that i's own top list did NOT already cover,
// atomically add sim_ij * x[j] into nbr[i] and sim_ij into cnt[i].
// ---------------------------------------------------------------------------
__global__ __launch_bounds__(256) void scatter_kernel(
    const float* __restrict__ x, const float* __restrict__ topv,
    const int* __restrict__ topi, float* __restrict__ nbr,
    float* __restrict__ cnt) {
  const int r = blockIdx.x;
  const int b = r / NS;
  const int j = r - b * NS;
  const int d = threadIdx.x;
  const float xv = x[(size_t)r * ND + d];
#pragma unroll
  for (int s = 0; s < 4; ++s) {
    const int i = topi[(size_t)r * 4 + s];
    if (i == j) continue;
    const int g = b * NS + i;
    bool dup = false;
#pragma unroll
    for (int u = 0; u < 4; ++u)
      if (topi[(size_t)g * 4 + u] == j) dup = true;
    if (dup) continue;
    const float v = topv[(size_t)r * 4 + s];
    atomicAdd(&nbr[(size_t)g * ND + d], v * xv);
    if (d == 0) atomicAdd(&cnt[g], v);
  }
}

// ---------------------------------------------------------------------------
// Kernel 6: out = LN(x + [x || nbr/max(cnt,1)] @ W^T + bias) * gamma + beta
// Block = 512 threads = 16 waves; each block owns 16 rows.
//  phase 1: stage combined [16 x 512] panel in LDS (stride 520, bank-safe)
//  phase 2: wave w -> 16x16 out tile (cols w*16..), K-loop 512 via f32 WMMA
//  phase 3: y = x + out + bias into LDS [16 x 264]
//  phase 4: per-row LayerNorm via wave32 shfl reductions
// ---------------------------------------------------------------------------
__global__ __launch_bounds__(512) void final_kernel(
    const float* __restrict__ x, const float* __restrict__ W,
    const float* __restrict__ bias, const float* __restrict__ gamma,
    const float* __restrict__ beta, const float* __restrict__ nbr,
    const float* __restrict__ cnt, float* __restrict__ out) {
  __shared__ __align__(16) float smem[16 * 520];  // 33280 B
  const int tid = threadIdx.x;
  const int r0 = blockIdx.x * 16;

  // phase 1: stage combined = [x | nbr_mean]
  for (int idx = tid; idx < 16 * NF; idx += 512) {
    const int rr = idx >> 9;        // / 512
    const int f = idx & (NF - 1);   // % 512
    const int r = r0 + rr;
    float val;
    if (f < ND) {
      val = x[(size_t)r * ND + f];
    } else {
      const float c = fmaxf(cnt[r], 1.0f);
      val = nbr[(size_t)r * ND + (f - ND)] / c;
    }
    smem[rr * 520 + f] = val;
  }
  __syncthreads();

  // phase 2: WMMA GEMM, A from LDS, B (= W rows) from global (L2 resident)
  const int wave = tid >> 5;
  const int lane = tid & 31;
  const int lrow = lane & 15;
  const int half = lane >> 4;
  const int n0 = wave * 16;
  const float* wrow = W + (size_t)(n0 + lrow) * NF + 2 * half;
  const float* aP = smem + lrow * 520 + 2 * half;

  v8f acc = {};
  for (int k = 0; k < NF; k += 4) {
    v2f a = *(const v2f*)(aP + k);     // ds_load_b64
    v2f b = *(const v2f*)(wrow + k);
    acc = __builtin_amdgcn_wmma_f32_16x16x4_f32(
        false, a, false, b, (short)0, acc, false, false);
  }
  __syncthreads();  // everyone done reading combined panel

  // phase 3: y = x + out + bias -> LDS [16 x 264]
  const int col = n0 + lrow;
  const float bcol = bias[col];
#pragma unroll
  for (int v = 0; v < 8; ++v) {
    const int mi = v + 8 * half;
    const float y = x[(size_t)(r0 + mi) * ND + col] + acc[v] + bcol;
    smem[mi * 264 + col] = y;
  }
  __syncthreads();

  // phase 4: LayerNorm; one wave (32 lanes) per row, 8 cols per lane
  const int row = tid >> 5;
  const int l = tid & 31;
  float s = 0.f, sq = 0.f;
#pragma unroll
  for (int q = 0; q < 8; ++q) {
    const float yv = smem[row * 264 + l + 32 * q];
    s += yv;
    sq += yv * yv;
  }
#pragma unroll
  for (int m = 16; m > 0; m >>= 1) {
    s += __shfl_xor(s, m, 32);
    sq += __shfl_xor(sq, m, 32);
  }
  const float mu = s * (1.0f / ND);
  const float var = sq * (1.0f / ND) - mu * mu;
  const float rs = rsqrtf(var + 1e-5f);
#pragma unroll
  for (int q = 0; q < 8; ++q) {
    const int c = l + 32 * q;
    const float yv = smem[row * 264 + c];
    out[(size_t)(r0 + row) * ND + c] = (yv - mu) * rs * gamma[c] + beta[c];
  }
}

// ---------------------------------------------------------------------------
// Host launcher
// ---------------------------------------------------------------------------
extern "C" void kernel_launch(void* const* d_in, const int* in_sizes, int n_in,
                              void* d_out, int out_size, void* d_ws,
                              size_t ws_size, hipStream_t stream) {
  const float* x = (const float*)d_in[0];      // [B,S,D]
  const float* W = (const float*)d_in[1];      // [D, 2D]
  const float* bias = (const float*)d_in[2];   // [D]
  const float* gamma = (const float*)d_in[3];  // [D]
  const float* beta = (const float*)d_in[4];   // [D]
  float* out = (float*)d_out;                  // [B,S,D]

  // workspace carve-up (floats): nrm | sim(1 batch) | topv | topi | nbr | cnt
  float* nrm = (float*)d_ws;                       // B*S*D
  float* sim = nrm + (size_t)NB * NS * ND;         // S*S (per-batch, reused)
  float* topv = sim + (size_t)NS * NS;             // B*S*4
  int* topi = (int*)(topv + (size_t)NB * NS * 4);  // B*S*4
  float* nbr = (float*)(topi + (size_t)NB * NS * 4);  // B*S*D
  float* cnt = nbr + (size_t)NB * NS * ND;         // B*S

  normalize_kernel<<<NB * NS, 256, 0, stream>>>(x, nrm);

  for (int b = 0; b < NB; ++b) {
    sim_kernel<<<dim3(NS / 32, NS / 256), 256, 0, stream>>>(
        nrm + (size_t)b * NS * ND, sim);
    topk_kernel<<<NS / 256, 256, 0, stream>>>(sim, topv, topi, b * NS);
  }

  gather_kernel<<<NB * NS, 256, 0, stream>>>(x, topv, topi, nbr, cnt);
  scatter_kernel<<<NB * NS, 256, 0, stream>>>(x, topv, topi, nbr, cnt);
  final_kernel<<<NB * NS / 16, 512, 0, stream>>>(x, W, bias, gamma, beta, nbr,
                                                 cnt, out);
}